// LexionAdapter_31817117729323
// MI455X (gfx1250) — compile-verified
//
#include <hip/hip_runtime.h>
#include <math.h>

typedef __attribute__((ext_vector_type(2))) float v2f;
typedef __attribute__((ext_vector_type(8))) float v8f;
typedef __attribute__((ext_vector_type(4))) unsigned int v4u;
typedef __attribute__((ext_vector_type(8))) int v8i;
typedef __attribute__((ext_vector_type(4))) int v4i;

#define WMMA_F32(a, b, c) \
    __builtin_amdgcn_wmma_f32_16x16x4_f32(false, (a), false, (b), (short)0, (c), false, false)

constexpr int Dh = 768;   // hidden size
constexpr int Ew = 200;   // word2vec embed size
constexpr int Wn = 8;     // words per char
constexpr int S  = 770;   // padded LDS row stride in floats (770 % 64 == 2 -> conflict-free)

// ---------------------------------------------------------------------------
// Mega-kernel: block of 256 threads (8 waves) handles 16 word-rows = 2 char rows.
//   Phase 0: TDM tensor_load_to_lds of the we tile [16 x 200] f32 -> LDS
//   Phase 1: wo1[16,768] = tanh(we @ Wt^T + bt)            (WMMA, K=200, A from LDS)
//   Phase 2: wo [16,768] = wo1 @ Ww^T + bw                 (WMMA, K=768, A from LDS)
//   Phase 3: scores = q.wo, softmax(W=8), weighted sum + residual + LayerNorm
// wo1/wo never touch HBM. LDS: 48.1KB tile + 12.8KB aux (we tile / reductions).
// ---------------------------------------------------------------------------
__global__ __launch_bounds__(256) void fused_word_attn(
    const float* __restrict__ we,     // [M1, E]
    const float* __restrict__ Wt,     // [D, E]
    const float* __restrict__ bt,     // [D]
    const float* __restrict__ Ww,     // [D, D]
    const float* __restrict__ bw,     // [D]
    const float* __restrict__ q,      // [M2, D]
    const float* __restrict__ layer,  // [M2, D]
    const float* __restrict__ gamma,  // [D]
    const float* __restrict__ beta,   // [D]
    float* __restrict__ out)          // [M2, D]
{
    __shared__ float s_wo[16 * S];    // 49280 B: wo1, then overwritten with wo
    __shared__ float s_aux[3200];     // 12800 B: phase1 = we tile [16][200]; phase3 = reductions

    const int t    = threadIdx.x;
    const int lane = t & 31;
    const int wv   = t >> 5;     // wave 0..7
    const int mr   = lane & 15;  // row (A) / col (B,D) within 16
    const int kh   = lane >> 4;  // k-half select

    const int m0 = blockIdx.x * 16;          // word-row base
    const int nb = wv * 96 + mr;             // this lane's base output column

    // ---- Phase 0: TDM DMA of the we tile into LDS (wave 0 issues, ISA 7 & 8) ----
    if (wv == 0) {
        const unsigned lds_addr = (unsigned)(size_t)(&s_aux[0]);
        const unsigned long long ga =
            (unsigned long long)(size_t)(we + (size_t)m0 * Ew);
        // D# group 0: count=1 | lds_addr | global_addr[56:0] | type=2
        v4u g0 = { 1u,
                   lds_addr,
                   (unsigned)(ga & 0xFFFFFFFFu),
                   (unsigned)((ga >> 32) & 0x01FFFFFFu) | 0x80000000u };
        // D# group 1: data_size=4B; tensor_dim0=200; tensor_dim1=65536;
        //             tile_dim0=200; tile_dim1=16; tensor_dim0_stride=200
        v8i g1 = { (int)0x00020000,             // [17:16] data_size=2 (4B)
                   (int)(200u << 16),           // [63:48] tensor_dim0 lo16 = 200
                   (int)0,                      // dim0 hi / tensor_dim1 lo16 = 0
                   (int)((200u << 16) | 1u),    // [127:112] tile_dim0=200, tensor_dim1 hi16=1
                   (int)16,                     // [143:128] tile_dim1=16 (tile_dim2=0)
                   (int)200,                    // tensor_dim0_stride lo32 = 200
                   0, 0 };
        v4i z4 = { 0, 0, 0, 0 };
        v8i z8 = { 0, 0, 0, 0, 0, 0, 0, 0 };
        __builtin_amdgcn_tensor_load_to_lds(g0, g1, z4, z4, z8, 0);
        __builtin_amdgcn_s_wait_tensorcnt((short)0);
    }
    // warm L2 for phase-3 operands while we compute
    {
        const float* q0 = q + (size_t)(blockIdx.x * 2) * Dh;
        const float* l0 = layer + (size_t)(blockIdx.x * 2) * Dh;
        __builtin_prefetch(q0 + lane * 32, 0, 1);          // 32 lanes x 128B = 4KB
        __builtin_prefetch(q0 + 1024 + lane * 16, 0, 1);   // tail
        __builtin_prefetch(l0 + lane * 32, 0, 1);
        __builtin_prefetch(l0 + 1024 + lane * 16, 0, 1);
    }
    __syncthreads();

    // ================= Phase 1: wo1 = tanh(we @ Wt^T + bt) =================
    {
        v8f acc[6];
#pragma unroll
        for (int j = 0; j < 6; ++j) acc[j] = (v8f){};
        const float* Aw = &s_aux[mr * Ew];   // we tile row, from LDS

        v2f a_c = *(const v2f*)(Aw + kh * 2);
        v2f b_c[6];
#pragma unroll
        for (int j = 0; j < 6; ++j)
            b_c[j] = *(const v2f*)(Wt + (size_t)(nb + j * 16) * Ew + kh * 2);

        for (int k = 0; k < Ew; k += 4) {
            v2f a_n = a_c, b_n[6];
#pragma unroll
            for (int j = 0; j < 6; ++j) b_n[j] = b_c[j];
            const int k2 = k + 4;
            if (k2 < Ew) {                    // prefetch next k-step fragments
                a_n = *(const v2f*)(Aw + k2 + kh * 2);
#pragma unroll
                for (int j = 0; j < 6; ++j)
                    b_n[j] = *(const v2f*)(Wt + (size_t)(nb + j * 16) * Ew + k2 + kh * 2);
            }
#pragma unroll
            for (int j = 0; j < 6; ++j) acc[j] = WMMA_F32(a_c, b_c[j], acc[j]);
            a_c = a_n;
#pragma unroll
            for (int j = 0; j < 6; ++j) b_c[j] = b_n[j];
        }
#pragma unroll
        for (int j = 0; j < 6; ++j) {
            const int n = nb + j * 16;
            const float bv = bt[n];
#pragma unroll
            for (int r = 0; r < 8; ++r)
                s_wo[(r + 8 * kh) * S + n] = tanhf(acc[j][r] + bv);
        }
    }
    __syncthreads();

    // ================= Phase 2: wo = wo1 @ Ww^T + bw (A from LDS) =================
    {
        v8f acc[6];
#pragma unroll
        for (int j = 0; j < 6; ++j) acc[j] = (v8f){};

        v2f a_c = *(const v2f*)(&s_wo[mr * S + kh * 2]);
        v2f b_c[6];
#pragma unroll
        for (int j = 0; j < 6; ++j)
            b_c[j] = *(const v2f*)(Ww + (size_t)(nb + j * 16) * Dh + kh * 2);

        for (int k = 0; k < Dh; k += 4) {
            v2f a_n = a_c, b_n[6];
#pragma unroll
            for (int j = 0; j < 6; ++j) b_n[j] = b_c[j];
            const int k2 = k + 4;
            if (k2 < Dh) {
                a_n = *(const v2f*)(&s_wo[mr * S + k2 + kh * 2]);
#pragma unroll
                for (int j = 0; j < 6; ++j)
                    b_n[j] = *(const v2f*)(Ww + (size_t)(nb + j * 16) * Dh + k2 + kh * 2);
            }
#pragma unroll
            for (int j = 0; j < 6; ++j) acc[j] = WMMA_F32(a_c, b_c[j], acc[j]);
            a_c = a_n;
#pragma unroll
            for (int j = 0; j < 6; ++j) b_c[j] = b_n[j];
        }
        __syncthreads();   // all LDS reads of wo1 complete before overwrite
#pragma unroll
        for (int j = 0; j < 6; ++j) {
            const int n = nb + j * 16;
            const float bv = bw[n];
#pragma unroll
            for (int r = 0; r < 8; ++r)
                s_wo[(r + 8 * kh) * S + n] = acc[j][r] + bv;
        }
    }
    __syncthreads();

    // ================= Phase 3: attention + softmax + residual + LN =================
    // group g (128 threads) handles char row cr; its wo rows are LDS rows g*8..g*8+7
    float* s_red = s_aux;          // [2][8][128]: g*1024 + w*128 + tg
    float* s_sc  = s_aux + 2048;   // [2][8]:      g*8 + w

    const int g  = t >> 7;       // 0/1
    const int tg = t & 127;      // 0..127 within group
    const int cr = blockIdx.x * 2 + g;
    const float* wog = &s_wo[g * 8 * S];

    const float* qrow = q + (size_t)cr * Dh;
    float qv[6];
#pragma unroll
    for (int i = 0; i < 6; ++i) qv[i] = qrow[tg + 128 * i];
#pragma unroll
    for (int w = 0; w < Wn; ++w) {
        float p = 0.0f;
#pragma unroll
        for (int i = 0; i < 6; ++i) p += qv[i] * wog[w * S + tg + 128 * i];
        s_red[g * 1024 + w * 128 + tg] = p;
    }
    __syncthreads();
    {   // 8 waves reduce the 16 (g,w) score partial-arrays: wave wv -> g=wv>>2, w=2*(wv&3)+h
        const int rg = wv >> 2;
#pragma unroll
        for (int h = 0; h < 2; ++h) {
            const int w = (wv & 3) * 2 + h;
            float p = 0.0f;
#pragma unroll
            for (int i = 0; i < 4; ++i) p += s_red[rg * 1024 + w * 128 + lane + 32 * i];
#pragma unroll
            for (int off = 16; off; off >>= 1) p += __shfl_down(p, off, 32);
            if (lane == 0) s_sc[rg * 8 + w] = p;
        }
    }
    __syncthreads();

    float aw[Wn];
    {
        float mx = s_sc[g * 8 + 0];
#pragma unroll
        for (int w = 1; w < Wn; ++w) mx = fmaxf(mx, s_sc[g * 8 + w]);
        float ss = 0.0f;
#pragma unroll
        for (int w = 0; w < Wn; ++w) { aw[w] = expf(s_sc[g * 8 + w] - mx); ss += aw[w]; }
        const float inv = 1.0f / ss;
#pragma unroll
        for (int w = 0; w < Wn; ++w) aw[w] *= inv;
    }

    const float* lrow = layer + (size_t)cr * Dh;
    float f[6], psum = 0.0f, psq = 0.0f;
#pragma unroll
    for (int i = 0; i < 6; ++i) {
        const int d = tg + 128 * i;
        float a = lrow[d];
#pragma unroll
        for (int w = 0; w < Wn; ++w) a += aw[w] * wog[w * S + d];
        f[i] = a; psum += a; psq += a * a;
    }
    __syncthreads();
    s_red[g * 1024 + 0 * 128 + tg] = psum;
    s_red[g * 1024 + 1 * 128 + tg] = psq;
    __syncthreads();
    {   // waves {0,1} reduce sum/sq for g0; waves {4,5} for g1
        const int rg = wv >> 2;
        const int v  = wv & 3;
        if (v < 2) {
            float p = 0.0f;
#pragma unroll
            for (int i = 0; i < 4; ++i) p += s_red[rg * 1024 + v * 128 + lane + 32 * i];
#pragma unroll
            for (int off = 16; off; off >>= 1) p += __shfl_down(p, off, 32);
            if (lane == 0) s_sc[rg * 8 + v] = p;
        }
    }
    __syncthreads();
    const float mean = s_sc[g * 8 + 0] * (1.0f / Dh);
    const float var  = s_sc[g * 8 + 1] * (1.0f / Dh) - mean * mean;
    const float rstd = rsqrtf(var + 1e-12f);
    float* orow = out + (size_t)cr * Dh;
#pragma unroll
    for (int i = 0; i < 6; ++i) {
        const int d = tg + 128 * i;
        orow[d] = (f[i] - mean) * rstd * gamma[d] + beta[d];
    }
}

// ---------------------------------------------------------------------------
// q = layer @ attn_W   (B in [K,N] layout), double-buffered WMMA GEMM.
// Wave tile 16x64 (4 accumulators), block = 8 waves = 128 rows x 64 cols.
// ---------------------------------------------------------------------------
__global__ __launch_bounds__(256) void gemm_q(
    const float* __restrict__ A,   // [M, 768]
    const float* __restrict__ Bm,  // [768, 768]
    float* __restrict__ C)         // [M, 768]
{
    const int lane = threadIdx.x & 31;
    const int wv   = threadIdx.x >> 5;
    const int mr   = lane & 15;
    const int kh   = lane >> 4;

    const int m0 = blockIdx.y * 128 + wv * 16;
    const int n0 = blockIdx.x * 64;

    v8f acc[4];
#pragma unroll
    for (int j = 0; j < 4; ++j) acc[j] = (v8f){};

    const float* Arow = A + (size_t)(m0 + mr) * Dh;

    auto loadB = [&](int k, v2f* b) {
        const int kk = k + kh * 2;
        const float* Bb = Bm + (size_t)kk * Dh + n0 + mr;
#pragma unroll
        for (int j = 0; j < 4; ++j) { b[j].x = Bb[j * 16]; b[j].y = Bb[j * 16 + Dh]; }
    };

    v2f a_c = *(const v2f*)(Arow + kh * 2);
    v2f b_c[4];
    loadB(0, b_c);

    for (int k = 0; k < Dh; k += 4) {
        v2f a_n = a_c, b_n[4];
#pragma unroll
        for (int j = 0; j < 4; ++j) b_n[j] = b_c[j];
        const int k2 = k + 4;
        if (k2 < Dh) {
            a_n = *(const v2f*)(Arow + k2 + kh * 2);
            loadB(k2, b_n);
        }
#pragma unroll
        for (int j = 0; j < 4; ++j) acc[j] = WMMA_F32(a_c, b_c[j], acc[j]);
        a_c = a_n;
#pragma unroll
        for (int j = 0; j < 4; ++j) b_c[j] = b_n[j];
    }

#pragma unroll
    for (int j = 0; j < 4; ++j) {
        const int n = n0 + j * 16 + mr;
#pragma unroll
        for (int r = 0; r < 8; ++r)
            C[(size_t)(m0 + r + 8 * kh) * Dh + n] = acc[j][r];
    }
}

// ---------------------------------------------------------------------------
extern "C" void kernel_launch(void* const* d_in, const int* in_sizes, int n_in,
                              void* d_out, int out_size, void* d_ws, size_t ws_size,
                              hipStream_t stream) {
    const float* we    = (const float*)d_in[0];  // [B,L,W,E]
    const float* layer = (const float*)d_in[1];  // [B,L,D]
    const float* Wt    = (const float*)d_in[2];  // [D,E]
    const float* bt    = (const float*)d_in[3];  // [D]
    const float* Ww    = (const float*)d_in[4];  // [D,D]
    const float* bw    = (const float*)d_in[5];  // [D]
    const float* attnW = (const float*)d_in[6];  // [D,D]
    const float* gamma = (const float*)d_in[7];  // [D]
    const float* beta  = (const float*)d_in[8];  // [D]
    float* out = (float*)d_out;

    constexpr int B = 16, L = 512;
    constexpr int M1 = B * L * Wn;   // 65536 word rows
    constexpr int M2 = B * L;        // 8192  char rows

    float* qb = (float*)d_ws;        // [M2, D] = 25 MB workspace

    // q = layer @ attn_W (independent; launch first)
    gemm_q<<<dim3(Dh / 64, M2 / 128), dim3(256), 0, stream>>>(layer, attnW, qb);

    // fused wo1 -> wo -> attention -> softmax -> residual -> layernorm
    fused_word_attn<<<dim3(M1 / 16), dim3(256), 0, stream>>>(
        we, Wt, bt, Ww, bw, qb, layer, gamma, beta, out);
}